// GCN_52355651338663
// MI455X (gfx1250) — compile-verified
//
#include <hip/hip_runtime.h>
#include <hip/hip_bf16.h>

typedef float v2f __attribute__((ext_vector_type(2)));
typedef float v8f __attribute__((ext_vector_type(8)));

#define HIDDEN 64

// ---------------- utility ----------------
__global__ void zero_f32(float* __restrict__ p, int n) {
    int i = blockIdx.x * blockDim.x + threadIdx.x;
    int stride = gridDim.x * blockDim.x;
    for (; i < n; i += stride) p[i] = 0.0f;
}

__global__ void init_deg(float* __restrict__ deg, int n) {
    int i = blockIdx.x * blockDim.x + threadIdx.x;
    if (i < n) deg[i] = 1.0f;   // self-loop
}

__global__ void deg_acc(float* __restrict__ deg, const int* __restrict__ dst, int e) {
    int i = blockIdx.x * blockDim.x + threadIdx.x;
    if (i < e) atomicAdd(&deg[dst[i]], 1.0f);
}

__global__ void deg_rsqrt(float* __restrict__ deg, int n) {
    int i = blockIdx.x * blockDim.x + threadIdx.x;
    if (i < n) deg[i] = rsqrtf(fmaxf(deg[i], 1.0f));   // in-place -> dinv
}

// per-edge symmetric norm, computed once and reused for all 3 layers
__global__ void edge_norm(const int* __restrict__ src, const int* __restrict__ dst,
                          const float* __restrict__ dinv, float* __restrict__ norm, int e) {
    int i = blockIdx.x * blockDim.x + threadIdx.x;
    if (i < e) norm[i] = dinv[src[i]] * dinv[dst[i]];
}

// ---------------- layer-1 GEMM: T = x(N x 9) @ W1(9 x 64) ----------------
__global__ __launch_bounds__(256) void gemm_in(const float* __restrict__ x,
                                               const float* __restrict__ W,
                                               float* __restrict__ T, int n) {
    int t = blockIdx.x * blockDim.x + threadIdx.x;
    int i = t >> 6;
    int c = t & 63;
    if (i >= n) return;
    const float* xr = x + (size_t)i * 9;
    float acc = 0.0f;
#pragma unroll
    for (int k = 0; k < 9; ++k) acc += xr[k] * W[k * 64 + c];
    T[(size_t)i * 64 + c] = acc;
}

// ---------------- WMMA GEMM: T = H(N x 64) @ W(64 x 64), fp32 ----------------
// One wave computes a 16x64 output tile via 4 accumulators of V_WMMA_F32_16X16X4_F32.
// W staged in LDS in paired-K layout: Ws2[k/2][col] = {W[k][col], W[k+1][col]},
// so every B fragment is a single contiguous ds_load_b64.
// Requires n to be a multiple of 16 (N=100000 is).
__global__ __launch_bounds__(256) void gemm_wmma(const float* __restrict__ H,
                                                 const float* __restrict__ W,
                                                 float* __restrict__ T, int n) {
    __shared__ float Ws[64 * 64];
    for (int i = threadIdx.x; i < 64 * 64; i += 256) {
        int k = i >> 6;
        int c = i & 63;
        // paired-K: element (k,c) -> pair index k/2, slot k&1
        Ws[((((k >> 1) << 6) + c) << 1) + (k & 1)] = W[i];
    }
    __syncthreads();

    int wave = threadIdx.x >> 5;
    int lane = threadIdx.x & 31;
    int tile_m = (blockIdx.x * 8 + wave) * 16;
    if (tile_m + 16 > n) return;   // wave-uniform exit; EXEC stays all-ones for WMMA

    int r   = lane & 15;            // row within tile (A) / column within n-tile (B,D)
    int khi = (lane >> 4) << 1;     // 0 for lanes 0-15, 2 for lanes 16-31
    int kpo = lane >> 4;            // pair-index offset: 0 or 1

    const float* Arow = H + (size_t)(tile_m + r) * 64;
    const v2f* Wp = (const v2f*)Ws;

    v8f acc0 = {}, acc1 = {}, acc2 = {}, acc3 = {};

    for (int k = 0; k < 64; k += 4) {
        // A fragment (16x4 f32): lane<16 holds K=k,k+1 ; lane>=16 holds K=k+2,k+3
        v2f a = *(const v2f*)(Arow + k + khi);

        // B fragments: pair (K=k+khi, K=k+khi+1) at column r + 16*ntile
        int kp = ((k >> 1) + kpo) << 6;
        v2f b0 = Wp[kp + r +  0];
        v2f b1 = Wp[kp + r + 16];
        v2f b2 = Wp[kp + r + 32];
        v2f b3 = Wp[kp + r + 48];

        acc0 = __builtin_amdgcn_wmma_f32_16x16x4_f32(false, a, false, b0, (short)0, acc0, false, false);
        acc1 = __builtin_amdgcn_wmma_f32_16x16x4_f32(false, a, false, b1, (short)0, acc1, false, false);
        acc2 = __builtin_amdgcn_wmma_f32_16x16x4_f32(false, a, false, b2, (short)0, acc2, false, false);
        acc3 = __builtin_amdgcn_wmma_f32_16x16x4_f32(false, a, false, b3, (short)0, acc3, false, false);
    }

    // D layout: VGPR v -> row tile_m + v + (lane>=16 ? 8 : 0), col = lane&15 within n-tile
    int mofs = (lane >> 4) << 3;
#pragma unroll
    for (int v = 0; v < 8; ++v) {
        float* out = T + (size_t)(tile_m + v + mofs) * 64 + r;
        out[0]  = acc0[v];
        out[16] = acc1[v];
        out[32] = acc2[v];
        out[48] = acc3[v];
    }
}

// ---------------- edge scatter: Z[dst] += T[src] * norm[e] ----------------
// 16 threads per edge, float4 per thread (L2-resident atomics).
__global__ __launch_bounds__(256) void scatter_edges(const float* __restrict__ T,
                                                     const int* __restrict__ src,
                                                     const int* __restrict__ dst,
                                                     const float* __restrict__ norm,
                                                     float* __restrict__ Z, int e) {
    int t = blockIdx.x * blockDim.x + threadIdx.x;
    int ed = t >> 4;
    if (ed >= e) return;
    int f4 = (t & 15) << 2;
    int s = src[ed];
    int d = dst[ed];
    float nrm = norm[ed];
    const float4 v = *(const float4*)(T + (size_t)s * 64 + f4);
    float* zp = Z + (size_t)d * 64 + f4;
    atomicAdd(zp + 0, v.x * nrm);
    atomicAdd(zp + 1, v.y * nrm);
    atomicAdd(zp + 2, v.z * nrm);
    atomicAdd(zp + 3, v.w * nrm);
}

// ---------------- finalize: H = relu(Z + T*dinv^2 + b) ----------------
__global__ __launch_bounds__(256) void finalize_layer(const float* __restrict__ Z,
                                                      const float* __restrict__ T,
                                                      const float* __restrict__ dinv,
                                                      const float* __restrict__ b,
                                                      float* __restrict__ H, int n) {
    int t = blockIdx.x * blockDim.x + threadIdx.x;
    int i = t >> 6;
    int c = t & 63;
    if (i >= n) return;
    float d = dinv[i];
    float v = Z[(size_t)i * 64 + c] + T[(size_t)i * 64 + c] * d * d + b[c];
    H[(size_t)i * 64 + c] = fmaxf(v, 0.0f);
}

// ---------------- mean pool accumulate ----------------
__global__ __launch_bounds__(256) void pool_acc(const float* __restrict__ H,
                                                const int* __restrict__ batch,
                                                float* __restrict__ pooled,
                                                float* __restrict__ cnt, int n) {
    int t = blockIdx.x * blockDim.x + threadIdx.x;
    int i = t >> 4;
    if (i >= n) return;
    int f4 = (t & 15) << 2;
    int g = batch[i];
    const float4 v = *(const float4*)(H + (size_t)i * 64 + f4);
    float* pp = pooled + (size_t)g * 64 + f4;
    atomicAdd(pp + 0, v.x);
    atomicAdd(pp + 1, v.y);
    atomicAdd(pp + 2, v.z);
    atomicAdd(pp + 3, v.w);
    if (f4 == 0) atomicAdd(&cnt[g], 1.0f);
}

// ---------------- MLP head: one block (64 thr) per graph ----------------
__global__ __launch_bounds__(64) void mlp_head(const float* __restrict__ pooled,
                                               const float* __restrict__ cnt,
                                               const float* __restrict__ u,
                                               const float* __restrict__ A1, const float* __restrict__ c1,
                                               const float* __restrict__ A2, const float* __restrict__ c2,
                                               const float* __restrict__ A3, const float* __restrict__ c3,
                                               const float* __restrict__ A4, const float* __restrict__ c4,
                                               const float* __restrict__ A5, const float* __restrict__ c5,
                                               float* __restrict__ out) {
    int g = blockIdx.x;
    int t = threadIdx.x;
    __shared__ float z0[68], z1[50], z2[30], z3[20], z4[5];

    float invc = 1.0f / fmaxf(cnt[g], 1.0f);
    if (t < 64) z0[t] = pooled[(size_t)g * 64 + t] * invc;
    if (t < 4)  z0[64 + t] = u[(size_t)g * 4 + t];
    __syncthreads();

    if (t < 50) {
        float a = c1[t];
        for (int k = 0; k < 68; ++k) a += z0[k] * A1[k * 50 + t];
        z1[t] = fmaxf(a, 0.0f);
    }
    __syncthreads();
    if (t < 30) {
        float a = c2[t];
        for (int k = 0; k < 50; ++k) a += z1[k] * A2[k * 30 + t];
        z2[t] = fmaxf(a, 0.0f);
    }
    __syncthreads();
    if (t < 20) {
        float a = c3[t];
        for (int k = 0; k < 30; ++k) a += z2[k] * A3[k * 20 + t];
        z3[t] = fmaxf(a, 0.0f);
    }
    __syncthreads();
    if (t < 5) {
        float a = c4[t];
        for (int k = 0; k < 20; ++k) a += z3[k] * A4[k * 5 + t];
        z4[t] = fmaxf(a, 0.0f);
    }
    __syncthreads();
    if (t == 0) {
        float a = c5[0];
        for (int k = 0; k < 5; ++k) a += z4[k] * A5[k];
        out[g] = fmaxf(a, 0.0f);
    }
}

// ---------------- launcher ----------------
extern "C" void kernel_launch(void* const* d_in, const int* in_sizes, int n_in,
                              void* d_out, int out_size, void* d_ws, size_t ws_size,
                              hipStream_t stream) {
    const float* x    = (const float*)d_in[0];
    const int*   eidx = (const int*)  d_in[1];
    const int*   batch= (const int*)  d_in[2];
    const float* u    = (const float*)d_in[3];
    const float* W1 = (const float*)d_in[4];  const float* b1 = (const float*)d_in[5];
    const float* W2 = (const float*)d_in[6];  const float* b2 = (const float*)d_in[7];
    const float* W3 = (const float*)d_in[8];  const float* b3 = (const float*)d_in[9];
    const float* A1 = (const float*)d_in[10]; const float* c1 = (const float*)d_in[11];
    const float* A2 = (const float*)d_in[12]; const float* c2 = (const float*)d_in[13];
    const float* A3 = (const float*)d_in[14]; const float* c3 = (const float*)d_in[15];
    const float* A4 = (const float*)d_in[16]; const float* c4 = (const float*)d_in[17];
    const float* A5 = (const float*)d_in[18]; const float* c5 = (const float*)d_in[19];
    float* out = (float*)d_out;

    const int N = in_sizes[0] / 9;
    const int E = in_sizes[1] / 2;
    const int G = in_sizes[3] / 4;
    const int* src = eidx;
    const int* dst = eidx + E;

    // workspace layout (256B aligned)
    auto alignup = [](size_t v) { return (v + 255) & ~(size_t)255; };
    char* w = (char*)d_ws;
    size_t off = 0;
    float* dinv   = (float*)(w + off); off += alignup((size_t)N * 4);
    float* norm   = (float*)(w + off); off += alignup((size_t)E * 4);
    float* T      = (float*)(w + off); off += alignup((size_t)N * HIDDEN * 4);
    float* Z      = (float*)(w + off); off += alignup((size_t)N * HIDDEN * 4);
    float* H      = (float*)(w + off); off += alignup((size_t)N * HIDDEN * 4);
    float* pooled = (float*)(w + off); off += alignup((size_t)G * HIDDEN * 4);
    float* cnt    = (float*)(w + off); off += alignup((size_t)G * 4);
    (void)ws_size; (void)n_in; (void)out_size;

    const int NT = 256;
    const int nfeat = N * HIDDEN;
    int gN    = (N + NT - 1) / NT;
    int gE    = (E + NT - 1) / NT;
    int gNF   = (nfeat + NT - 1) / NT;
    int gE16  = (int)(((long long)E * 16 + NT - 1) / NT);
    int gN16  = (int)(((long long)N * 16 + NT - 1) / NT);
    int gWmma = ((N / 16) + 7) / 8;   // 8 waves/block, one 16-row tile per wave

    // degrees -> dinv (in place) -> per-edge norm
    init_deg<<<gN, NT, 0, stream>>>(dinv, N);
    deg_acc<<<gE, NT, 0, stream>>>(dinv, dst, E);
    deg_rsqrt<<<gN, NT, 0, stream>>>(dinv, N);
    edge_norm<<<gE, NT, 0, stream>>>(src, dst, dinv, norm, E);

    // ---- layer 1: K=9 scalar GEMM ----
    gemm_in<<<gNF, NT, 0, stream>>>(x, W1, T, N);
    zero_f32<<<gNF, NT, 0, stream>>>(Z, nfeat);
    scatter_edges<<<gE16, NT, 0, stream>>>(T, src, dst, norm, Z, E);
    finalize_layer<<<gNF, NT, 0, stream>>>(Z, T, dinv, b1, H, N);

    // ---- layer 2: fp32 WMMA GEMM ----
    gemm_wmma<<<gWmma, NT, 0, stream>>>(H, W2, T, N);
    zero_f32<<<gNF, NT, 0, stream>>>(Z, nfeat);
    scatter_edges<<<gE16, NT, 0, stream>>>(T, src, dst, norm, Z, E);
    finalize_layer<<<gNF, NT, 0, stream>>>(Z, T, dinv, b2, H, N);

    // ---- layer 3: fp32 WMMA GEMM ----
    gemm_wmma<<<gWmma, NT, 0, stream>>>(H, W3, T, N);
    zero_f32<<<gNF, NT, 0, stream>>>(Z, nfeat);
    scatter_edges<<<gE16, NT, 0, stream>>>(T, src, dst, norm, Z, E);
    finalize_layer<<<gNF, NT, 0, stream>>>(Z, T, dinv, b3, H, N);

    // ---- mean pool + MLP head ----
    zero_f32<<<(G * HIDDEN + NT - 1) / NT, NT, 0, stream>>>(pooled, G * HIDDEN);
    zero_f32<<<(G + NT - 1) / NT, NT, 0, stream>>>(cnt, G);
    pool_acc<<<gN16, NT, 0, stream>>>(H, batch, pooled, cnt, N);
    mlp_head<<<G, 64, 0, stream>>>(pooled, cnt, u,
                                   A1, c1, A2, c2, A3, c3, A4, c4, A5, c5, out);
}